// VoronoiWatershedLoss_18107582120047
// MI455X (gfx1250) — compile-verified
//
#include <hip/hip_runtime.h>
#include <math.h>

#define JN 128
#define HH 512        // half-res
#define WH 512
#define HF 1024       // full-res
#define WF 1024
#define EPSV 1e-7f
#define DEF_SIGMA 4096.0f

typedef __attribute__((ext_vector_type(2))) float v2f;
typedef __attribute__((ext_vector_type(8))) float v8f;

// ---------------------------------------------------------------------------
// Kernel 1: per-Gaussian prep. Closed-form eigh of 2x2 SPD sigma, rescaled
// covariance sg = V diag(Lsc) V^T / D^2, inverse, and the 8 quadratic-form
// coefficients so that maha(pixel,j) = sum_k C[j][k] * F[pixel][k] with
// F = [x^2, x*y, y^2, x, y, 1, 0, 0].  Also zero-inits segment accumulators.
// ---------------------------------------------------------------------------
__global__ void vwl_prep(const float* __restrict__ mu,
                         const float* __restrict__ sigma,
                         float* __restrict__ coef,   // [JN][8]
                         float* __restrict__ vs,     // [JN][4] col0=(v0x,v0y), col1=(v1x,v1y)
                         float* __restrict__ ld,     // [JN][2] ascending eigenvalues
                         unsigned* __restrict__ instMax,  // [JN][2]
                         unsigned* __restrict__ instCnt)  // [JN]
{
    int t = threadIdx.x;
    if (t < JN) {
        float a = sigma[4*t + 0];
        float b = sigma[4*t + 1];
        float c = sigma[4*t + 3];
        float mean = 0.5f * (a + c);
        float hd   = 0.5f * (a - c);
        float s    = sqrtf(hd*hd + b*b);
        float l0 = mean - s;           // ascending, SPD so l0 >= 100
        float l1 = mean + s;
        // eigenvector for l1: pick the better-conditioned null-space row
        float d1a = l1 - a, d1c = l1 - c;
        float e1x, e1y;
        if (fabsf(d1c) >= fabsf(d1a)) { e1x = d1c; e1y = b; }
        else                          { e1x = b;   e1y = d1a; }
        float nrm = sqrtf(e1x*e1x + e1y*e1y);
        if (nrm > 0.0f) { e1x /= nrm; e1y /= nrm; }
        else            { e1x = 1.0f; e1y = 0.0f; }      // isotropic degenerate
        float e0x = -e1y, e0y = e1x;                      // orthogonal complement

        ld[2*t + 0] = l0;  ld[2*t + 1] = l1;
        vs[4*t + 0] = e0x; vs[4*t + 1] = e0y;
        vs[4*t + 2] = e1x; vs[4*t + 3] = e1y;

        // Lsc = Ld / sqrt(Ld0*Ld1) * DEFAULT_SIGMA ;  sg = V diag(Lsc) V^T / 4
        float gm  = sqrtf(l0 * l1);
        float ls0 = l0 / gm * DEF_SIGMA;
        float ls1 = l1 / gm * DEF_SIGMA;
        float sa = (ls0*e0x*e0x + ls1*e1x*e1x) * 0.25f;
        float sb = (ls0*e0x*e0y + ls1*e1x*e1y) * 0.25f;
        float sc = (ls0*e0y*e0y + ls1*e1y*e1y) * 0.25f;
        float det = sa*sc - sb*sb;
        float ia =  sc / det;
        float ib = -sb / det;
        float ic =  sa / det;
        float mmx = rintf(mu[2*t + 0] * 0.5f);   // round-half-even == jnp.round
        float mmy = rintf(mu[2*t + 1] * 0.5f);

        coef[8*t + 0] = ia;
        coef[8*t + 1] = 2.0f * ib;
        coef[8*t + 2] = ic;
        coef[8*t + 3] = -2.0f * (ia*mmx + ib*mmy);
        coef[8*t + 4] = -2.0f * (ib*mmx + ic*mmy);
        coef[8*t + 5] = ia*mmx*mmx + 2.0f*ib*mmx*mmy + ic*mmy*mmy;
        coef[8*t + 6] = 0.0f;
        coef[8*t + 7] = 0.0f;

        instMax[2*t] = 0u; instMax[2*t + 1] = 0u; instCnt[t] = 0u;
    }
}

// ---------------------------------------------------------------------------
// Kernel 2: the hot GEMM.  maha[16 pixels x 16 gaussians] tiles via
// V_WMMA_F32_16X16X4_F32 (two chained K=4 ops, K padded 6->8), then
// g = exp(-0.5*maha), running max/argmax over all 8 gaussian tiles.
// A: M=gaussian rows (lane L supplies M=L&15; VGPR0/1 hold K0,1 for lanes
//    0-15 and K2,3 for lanes 16-31 per the CDNA5 f32 A-matrix layout).
// B: N=pixel cols (each lane supplies its own pixel's feature rows).
// D: lane L holds column N=L&15, rows M = v + (L>=16 ? 8 : 0), v=0..7.
// One shfl_xor(16) merges the two M-halves with argmax-first tie-breaking.
// ---------------------------------------------------------------------------
__global__ __launch_bounds__(256) void vwl_voronoi(
    const float* __restrict__ coef,
    float* __restrict__ val,    // [HH*WH]
    int*   __restrict__ vor)    // [HH*WH]
{
    const int lane  = threadIdx.x & 31;
    const int wave  = threadIdx.x >> 5;
    const int p     = blockIdx.x * 128 + wave * 16 + (lane & 15);
    const int r     = p >> 9;           // / WH
    const int c     = p & (WH - 1);

    // linspace(0, 512, 512) => step 512/511
    const float S = 512.0f / 511.0f;
    const float x = (float)c * S;
    const float y = (float)r * S;
    float F[8] = { x*x, x*y, y*y, x, y, 1.0f, 0.0f, 0.0f };

    const int k0 = (lane & 16) ? 2 : 0;
    v2f b0; b0.x = F[k0];     b0.y = F[k0 + 1];
    v2f b1; b1.x = F[k0 + 4]; b1.y = F[k0 + 5];

    float best_g = -1.0f;
    int   best_j = 0;

    for (int t = 0; t < 8; ++t) {
        const float* cj = coef + 8 * (t * 16 + (lane & 15));
        v2f a0; a0.x = cj[k0];     a0.y = cj[k0 + 1];
        v2f a1; a1.x = cj[k0 + 4]; a1.y = cj[k0 + 5];

        v8f acc = {0.f, 0.f, 0.f, 0.f, 0.f, 0.f, 0.f, 0.f};
        acc = __builtin_amdgcn_wmma_f32_16x16x4_f32(
                  false, a0, false, b0, (short)0, acc, false, false);
        acc = __builtin_amdgcn_wmma_f32_16x16x4_f32(
                  false, a1, false, b1, (short)0, acc, false, false);

        union { v8f v; float f[8]; } u; u.v = acc;
        const int jb = t * 16 + ((lane & 16) ? 8 : 0);
        #pragma unroll
        for (int v = 0; v < 8; ++v) {
            float g = expf(-0.5f * u.f[v]);
            if (g > best_g) { best_g = g; best_j = jb + v; }   // strict > => first max
        }
    }
    // merge the two M-halves (each lane pair holds the same pixel column)
    float og = __shfl_xor(best_g, 16);
    int   oj = __shfl_xor(best_j, 16);
    if (og > best_g || (og == best_g && oj < best_j)) { best_g = og; best_j = oj; }

    if ((lane & 16) == 0) {
        val[p] = best_g;
        vor[p] = best_j;
    }
}

// ---------------------------------------------------------------------------
// Kernel 3: full-res fused markers + segment reductions.
// vor_full = 2x nearest upsample; val_full = align-corners bilinear;
// ridges = (3x3 ones/-8 laplacian of vor_full != 0), zero-padded;
// marker cascade; per-instance max|rot| and count via LDS-privatized
// uint-bit atomicMax (|rot| >= 0 so float bits are order-preserving).
// ---------------------------------------------------------------------------
__global__ __launch_bounds__(256) void vwl_markers(
    const float* __restrict__ mu,
    const int*   __restrict__ label,
    const float* __restrict__ pos_thres,
    const float* __restrict__ neg_thres,
    const float* __restrict__ vs,
    const float* __restrict__ val,
    const int*   __restrict__ vor,
    unsigned* __restrict__ instMax,
    unsigned* __restrict__ instCnt)
{
    __shared__ unsigned sMax[2 * JN];
    __shared__ unsigned sCnt[JN];
    const int t = threadIdx.x;
    for (int i = t; i < JN; i += 256) { sMax[2*i] = 0u; sMax[2*i+1] = 0u; sCnt[i] = 0u; }
    __syncthreads();

    const int p = blockIdx.x * 256 + t;
    const int r = p >> 10;          // / WF
    const int c = p & (WF - 1);

    const int vc = vor[(r >> 1) * WH + (c >> 1)];

    // integer laplacian with zero padding: lap = sum(8 neighbors) - 8*center
    int ssum = 0;
    #pragma unroll
    for (int dr = -1; dr <= 1; ++dr) {
        #pragma unroll
        for (int dc = -1; dc <= 1; ++dc) {
            if (dr == 0 && dc == 0) continue;
            int rr = r + dr, cc = c + dc;
            if (rr >= 0 && rr < HF && cc >= 0 && cc < WF)
                ssum += vor[(rr >> 1) * WH + (cc >> 1)];
        }
    }
    const bool ridge = (ssum - 8 * vc) != 0;

    // bilinear (align-corners style): ry = r*511/1023
    float ry = ((float)r * 511.0f) / 1023.0f;
    float rx = ((float)c * 511.0f) / 1023.0f;
    int y0 = (int)floorf(ry); int y1 = (y0 + 1 < HH - 1) ? y0 + 1 : HH - 1;
    int x0 = (int)floorf(rx); int x1 = (x0 + 1 < WH - 1) ? x0 + 1 : WH - 1;
    float wy = ry - (float)y0;
    float wx = rx - (float)x0;
    float v00 = val[y0*WH + x0], v01 = val[y0*WH + x1];
    float v10 = val[y1*WH + x0], v11 = val[y1*WH + x1];
    float top = v00 * (1.0f - wx) + v01 * wx;
    float bot = v10 * (1.0f - wx) + v11 * wx;
    float vf  = top * (1.0f - wy) + bot * wy;

    const int cl = label[vc];
    int m = vc + 1;
    if (vf < pos_thres[cl]) m = 0;
    if (vf < neg_thres[cl]) m = JN + 1;
    if (ridge)              m = JN + 1;

    // only segments 1..J feed L_target
    if (m >= 1 && m <= JN) {
        const int inst = m - 1;
        float dx = (float)c - mu[2*inst + 0];
        float dy = (float)r - mu[2*inst + 1];
        float r0 = fabsf(vs[4*inst + 0] * dx + vs[4*inst + 1] * dy);
        float r1 = fabsf(vs[4*inst + 2] * dx + vs[4*inst + 3] * dy);
        atomicMax(&sMax[2*inst + 0], __float_as_uint(r0));
        atomicMax(&sMax[2*inst + 1], __float_as_uint(r1));
        atomicAdd(&sCnt[inst], 1u);
    }
    __syncthreads();
    for (int i = t; i < JN; i += 256) {
        if (sCnt[i]) {
            atomicMax(&instMax[2*i + 0], sMax[2*i + 0]);
            atomicMax(&instMax[2*i + 1], sMax[2*i + 1]);
            atomicAdd(&instCnt[i], sCnt[i]);
        }
    }
}

// ---------------------------------------------------------------------------
// Kernel 4: per-instance Wasserstein-ish eigenvalue loss + mean of the
// 122 smallest (= total minus the 6 largest, / 122).
// ---------------------------------------------------------------------------
__global__ void vwl_loss(const float* __restrict__ ld,
                         const unsigned* __restrict__ instMax,
                         const unsigned* __restrict__ instCnt,
                         float* __restrict__ out)
{
    __shared__ float ls[JN];
    const int j = threadIdx.x;
    if (j < JN) {
        float Lp0 = ld[2*j], Lp1 = ld[2*j + 1];
        unsigned cnt = instCnt[j];
        float m0 = __uint_as_float(instMax[2*j + 0]);
        float m1 = __uint_as_float(instMax[2*j + 1]);
        float Lt0 = cnt ? m0 * m0 : Lp0;
        float Lt1 = cnt ? m1 * m1 : Lp1;
        float whr = Lp0 + Lp1 + Lt0 + Lt1;
        float tr  = Lp0 * Lt0 + Lp1 * Lt1;
        float det_sqrt = sqrtf(fmaxf(Lp0 * Lp1 * Lt0 * Lt1, EPSV));
        whr = whr - 2.0f * sqrtf(fmaxf(tr + 2.0f * det_sqrt, EPSV));
        float dist  = sqrtf(fmaxf(whr, EPSV));
        float scale = 2.0f * fmaxf(sqrtf(fmaxf(sqrtf(fmaxf(det_sqrt, EPSV)), EPSV)), EPSV);
        dist /= scale;
        ls[j] = 1.0f - 1.0f / (1.0f + log1pf(dist));
    }
    __syncthreads();
    if (j == 0) {
        const int kkeep = 122;              // ceil(128 * 0.95)
        float total = 0.0f;
        for (int i = 0; i < JN; ++i) total += ls[i];
        for (int e = 0; e < JN - kkeep; ++e) {
            int bi = 0; float bv = ls[0];
            for (int i = 1; i < JN; ++i) if (ls[i] > bv) { bv = ls[i]; bi = i; }
            total -= bv; ls[bi] = -1e30f;
        }
        out[0] = total / (float)kkeep;      // LOSS_WEIGHT = 1.0
    }
}

// ---------------------------------------------------------------------------
extern "C" void kernel_launch(void* const* d_in, const int* in_sizes, int n_in,
                              void* d_out, int out_size, void* d_ws, size_t ws_size,
                              hipStream_t stream)
{
    const float* mu    = (const float*)d_in[0];   // (128,2)
    const float* sigma = (const float*)d_in[1];   // (128,2,2)
    const int*   label = (const int*)  d_in[2];   // (128,)
    // d_in[3] = image (3,1024,1024): unused by the forward pass
    const float* pos   = (const float*)d_in[4];   // (16,)
    const float* neg   = (const float*)d_in[5];   // (16,)

    float* ws   = (float*)d_ws;
    float* coef = ws;                       // 128*8
    float* vs   = coef + JN * 8;            // 128*4
    float* ld   = vs   + JN * 4;            // 128*2
    float* val  = ld   + JN * 2;            // 512*512
    int*   vor  = (int*)(val + HH * WH);    // 512*512
    unsigned* instMax = (unsigned*)(vor + HH * WH);  // 128*2
    unsigned* instCnt = instMax + 2 * JN;            // 128

    vwl_prep<<<1, 128, 0, stream>>>(mu, sigma, coef, vs, ld, instMax, instCnt);
    vwl_voronoi<<<(HH * WH) / 128, 256, 0, stream>>>(coef, val, vor);
    vwl_markers<<<(HF * WF) / 256, 256, 0, stream>>>(mu, label, pos, neg, vs,
                                                     val, vor, instMax, instCnt);
    vwl_loss<<<1, 128, 0, stream>>>(ld, instMax, instCnt, (float*)d_out);
}